// TreeAttentionSiblings_53541062312191
// MI455X (gfx1250) — compile-verified
//
#include <hip/hip_runtime.h>
#include <hip/hip_bf16.h>

typedef __attribute__((ext_vector_type(2))) float v2f;
typedef __attribute__((ext_vector_type(8))) float v8f;

#define NPTS   32768
#define DIM    96
#define DH     16
#define KNNK   16
#define NB     16
#define MPTS   2048
#define QTILES 128          // MPTS/16
#define W_K1   4            // waves per block, knn kernel
#define BN_BLOCKS 256
#define BN_CNT ((float)(NPTS * KNNK))

// ---------------------------------------------------------------------------
// Kernel 1: per-cloud KNN via WMMA f32 16x16x4.
//   A (16x4, candidates) = (-2x, -2y | -2z, |c|^2)
//   B (4x16, queries)    = ( x,   y  |  z,  1.0 )
//   => acc[m][q] = |c_m|^2 - 2 c_m.q  (== dist - |q|^2; |q|^2 is a per-query
//      constant so top-k selection is unchanged without it).
// One wave owns a 16-query tile; 128 WMMAs sweep all 2048 candidates.
// ---------------------------------------------------------------------------
__global__ __launch_bounds__(128) void knn_kernel(const float* __restrict__ p,
                                                  int* __restrict__ gidx) {
  __shared__ float4 pts[MPTS];           // x,y,z,|p|^2  (32 KB)
  __shared__ float  md[W_K1][16][33];    // per-lane 16 best keys (padded)
  __shared__ int    mi[W_K1][16][33];

  const int batch = blockIdx.x >> 5;     // 32 blocks per batch
  const int tblk  = blockIdx.x & 31;
  const int tid   = threadIdx.x;
  const int wave  = tid >> 5;
  const int lane  = tid & 31;
  const int q16   = lane & 15;
  const int half  = lane >> 4;
  const int base  = batch * MPTS;

  for (int i = tid; i < MPTS; i += 128) {
    float x = p[(base + i) * 3 + 0];
    float y = p[(base + i) * 3 + 1];
    float z = p[(base + i) * 3 + 2];
    pts[i] = make_float4(x, y, z, x * x + y * y + z * z);
  }
  __syncthreads();

  const int qt = tblk * W_K1 + wave;     // query tile 0..127
  const float4 qp = pts[qt * 16 + q16];
  v2f bmat;                              // lanes<16: K=0,1 ; lanes>=16: K=2,3
  bmat.x = (half == 0) ? qp.x : qp.z;
  bmat.y = (half == 0) ? qp.y : 1.0f;

  float* slotD = &md[wave][q16][half * 16];
  int*   slotI = &mi[wave][q16][half * 16];
#pragma unroll
  for (int s = 0; s < 16; ++s) { slotD[s] = __builtin_inff(); slotI[s] = 0; }
  float worst = __builtin_inff();
  int   wslot = 0;

#pragma unroll 2
  for (int ct = 0; ct < QTILES; ++ct) {
    const int cbase = ct * 16;
    const float4 cp = pts[cbase + q16];
    v2f amat;
    amat.x = (half == 0) ? (-2.0f * cp.x) : (-2.0f * cp.z);
    amat.y = (half == 0) ? (-2.0f * cp.y) : cp.w;
    v8f acc = {0.f, 0.f, 0.f, 0.f, 0.f, 0.f, 0.f, 0.f};
    acc = __builtin_amdgcn_wmma_f32_16x16x4_f32(false, amat, false, bmat,
                                                (short)0, acc, false, false);
    const int mBase = cbase + half * 8;  // C rows: lanes<16 -> M=v, lanes>=16 -> M=v+8
#pragma unroll
    for (int v = 0; v < 8; ++v) {
      const float d = acc[v];            // compare key straight from accumulator
      if (d < worst) {
        slotD[wslot] = d; slotI[wslot] = mBase + v;
        float w0 = slotD[0]; int s0 = 0;
#pragma unroll
        for (int s = 1; s < 16; ++s) {
          float ds = slotD[s];
          if (ds > w0) { w0 = ds; s0 = s; }
        }
        worst = w0; wslot = s0;
      }
    }
  }
  __syncthreads();

  // merge the two 16-candidate halves per query -> 16 nearest (order-free)
  if (lane < 16) {
    const int row = base + qt * 16 + lane;
    float* dd = &md[wave][lane][0];
    int*   ii = &mi[wave][lane][0];
    for (int t = 0; t < KNNK; ++t) {
      float best = dd[0]; int bj = 0;
      for (int j = 1; j < 32; ++j) {
        float v = dd[j];
        if (v < best) { best = v; bj = j; }
      }
      gidx[row * KNNK + t] = base + ii[bj];
      dd[bj] = __builtin_inff();
    }
  }
}

// ---------------------------------------------------------------------------
// Kernel 2: per-block partial sums for BatchNorm mean/var (deterministic)
// ---------------------------------------------------------------------------
__global__ __launch_bounds__(256) void bn_partial_kernel(const float* __restrict__ p,
                                                         const int* __restrict__ gidx,
                                                         const float* __restrict__ W1,
                                                         const float* __restrict__ b1,
                                                         float* __restrict__ partial) {
  __shared__ float red[256];
  float s[3] = {0.f, 0.f, 0.f}, ss[3] = {0.f, 0.f, 0.f};
  const int tid = threadIdx.x;
  const int rbase = blockIdx.x * 2048;
  for (int i = 0; i < 8; ++i) {
    int r = rbase + tid + i * 256;
    int n = r >> 4;
    int g = gidx[r];
    float dx = p[n * 3 + 0] - p[g * 3 + 0];
    float dy = p[n * 3 + 1] - p[g * 3 + 1];
    float dz = p[n * 3 + 2] - p[g * 3 + 2];
#pragma unroll
    for (int c = 0; c < 3; ++c) {
      float h = dx * W1[0 * 3 + c] + dy * W1[1 * 3 + c] + dz * W1[2 * 3 + c] + b1[c];
      s[c] += h; ss[c] += h * h;
    }
  }
#pragma unroll
  for (int c = 0; c < 6; ++c) {
    red[tid] = (c < 3) ? s[c] : ss[c - 3];
    __syncthreads();
    for (int off = 128; off > 0; off >>= 1) {
      if (tid < off) red[tid] += red[tid + off];
      __syncthreads();
    }
    if (tid == 0) partial[blockIdx.x * 6 + c] = red[0];
    __syncthreads();
  }
}

__global__ void bn_finalize_kernel(const float* __restrict__ partial,
                                   float* __restrict__ stats) {
  __shared__ float tot[6];
  const int t = threadIdx.x;
  if (t < 6) {
    float s = 0.f;
    for (int b = 0; b < BN_BLOCKS; ++b) s += partial[b * 6 + t];
    tot[t] = s;
  }
  __syncthreads();
  if (t < 3) {
    float mean = tot[t] / BN_CNT;
    float var  = tot[t + 3] / BN_CNT - mean * mean;
    stats[t]     = mean;
    stats[3 + t] = rsqrtf(var + 1e-5f);
  }
}

// ---------------------------------------------------------------------------
// Kernel 3: vector attention. One 96-thread block per point, thread = out dim.
// 16-lane shfl_xor reductions (wave32: heads align to half-waves).
// ---------------------------------------------------------------------------
__global__ __launch_bounds__(96) void attn_kernel(const float* __restrict__ p,
                                                  const float* __restrict__ Q,
                                                  const float* __restrict__ K,
                                                  const float* __restrict__ V,
                                                  const float* __restrict__ W1,
                                                  const float* __restrict__ b1,
                                                  const float* __restrict__ gamma,
                                                  const float* __restrict__ beta,
                                                  const float* __restrict__ W2,
                                                  const float* __restrict__ b2,
                                                  const int* __restrict__ gidx,
                                                  const float* __restrict__ stats,
                                                  float* __restrict__ out) {
  __shared__ int   gsh[KNNK];
  __shared__ float wsc[KNNK][8];
  const int n = blockIdx.x;
  const int t = threadIdx.x;           // 0..95 = output dim
  const int head = t >> 4;
  const int j = t & 15;

  if (t < KNNK) gsh[t] = gidx[n * KNNK + t];
  __syncthreads();

  const float qv = Q[n * DIM + t];
  const float px = p[n * 3 + 0], py = p[n * 3 + 1], pz = p[n * 3 + 2];
  const float m0 = stats[0], m1 = stats[1], m2 = stats[2];
  const float r0c = stats[3], r1c = stats[4], r2c = stats[5];
  const float w00 = W1[0], w01 = W1[1], w02 = W1[2];
  const float w10 = W1[3], w11 = W1[4], w12 = W1[5];
  const float w20 = W1[6], w21 = W1[7], w22 = W1[8];
  const float bb0 = b1[0], bb1 = b1[1], bb2 = b1[2];
  const float ga0 = gamma[0], ga1 = gamma[1], ga2 = gamma[2];
  const float be0 = beta[0], be1 = beta[1], be2 = beta[2];
  const float u0 = W2[0 * DH + j], u1 = W2[1 * DH + j], u2 = W2[2 * DH + j];
  const float b2j = b2[j];

  float vv[KNNK], ph[KNNK];
#pragma unroll
  for (int k = 0; k < KNNK; ++k) {
    const int g = gsh[k];
    const float kv = K[g * DIM + t];
    vv[k] = V[g * DIM + t];
    float prod = kv * qv;                       // per-head dot over dh=16 lanes
    prod += __shfl_xor(prod, 8, 32);
    prod += __shfl_xor(prod, 4, 32);
    prod += __shfl_xor(prod, 2, 32);
    prod += __shfl_xor(prod, 1, 32);
    const float dx = px - p[g * 3 + 0];
    const float dy = py - p[g * 3 + 1];
    const float dz = pz - p[g * 3 + 2];
    const float h0 = dx * w00 + dy * w10 + dz * w20 + bb0;
    const float h1 = dx * w01 + dy * w11 + dz * w21 + bb1;
    const float h2 = dx * w02 + dy * w12 + dz * w22 + bb2;
    const float e0 = fmaxf((h0 - m0) * r0c * ga0 + be0, 0.f);
    const float e1 = fmaxf((h1 - m1) * r1c * ga1 + be1, 0.f);
    const float e2 = fmaxf((h2 - m2) * r2c * ga2 + be2, 0.f);
    const float peh = e0 * u0 + e1 * u1 + e2 * u2 + b2j;
    ph[k] = peh;
    float sp = peh;                             // Sum_d peh (head-independent)
    sp += __shfl_xor(sp, 8, 32);
    sp += __shfl_xor(sp, 4, 32);
    sp += __shfl_xor(sp, 2, 32);
    sp += __shfl_xor(sp, 1, 32);
    const float wv = (prod + sp) * (1.0f / 64.0f);
    if (j == 0) wsc[k][head] = wv;
  }
  __syncthreads();

  float wreg[KNNK];
  float mx = -__builtin_inff();
#pragma unroll
  for (int k = 0; k < KNNK; ++k) { wreg[k] = wsc[k][head]; mx = fmaxf(mx, wreg[k]); }
  float denom = 0.f;
#pragma unroll
  for (int k = 0; k < KNNK; ++k) { float e = __expf(wreg[k] - mx); wreg[k] = e; denom += e; }
  float acc = 0.f;
#pragma unroll
  for (int k = 0; k < KNNK; ++k) acc += (vv[k] + ph[k]) * wreg[k];
  out[n * DIM + t] = acc / denom;
}

// ---------------------------------------------------------------------------
extern "C" void kernel_launch(void* const* d_in, const int* in_sizes, int n_in,
                              void* d_out, int out_size, void* d_ws, size_t ws_size,
                              hipStream_t stream) {
  const float* p     = (const float*)d_in[0];
  const float* Q     = (const float*)d_in[1];
  const float* K     = (const float*)d_in[2];
  const float* V     = (const float*)d_in[3];
  const float* W1    = (const float*)d_in[4];
  const float* b1    = (const float*)d_in[5];
  const float* gamma = (const float*)d_in[6];
  const float* beta  = (const float*)d_in[7];
  const float* W2    = (const float*)d_in[8];
  const float* b2    = (const float*)d_in[9];
  float* out = (float*)d_out;

  char* ws = (char*)d_ws;
  int*   gidx    = (int*)ws;                                        // N*16 ints
  float* partial = (float*)(ws + (size_t)NPTS * KNNK * 4);          // 256*6 f32
  float* stats   = (float*)(ws + (size_t)NPTS * KNNK * 4 + 8192);   // mean[3],rstd[3]

  knn_kernel<<<dim3(NB * 32), dim3(128), 0, stream>>>(p, gidx);
  bn_partial_kernel<<<dim3(BN_BLOCKS), dim3(256), 0, stream>>>(p, gidx, W1, b1, partial);
  bn_finalize_kernel<<<dim3(1), dim3(32), 0, stream>>>(partial, stats);
  attn_kernel<<<dim3(NPTS), dim3(96), 0, stream>>>(p, Q, K, V, W1, b1, gamma, beta,
                                                   W2, b2, gidx, stats, out);
}